// SparsePatternSeparator_78048145703037
// MI455X (gfx1250) — compile-verified
//
#include <hip/hip_runtime.h>
#include <math.h>

typedef float v2f __attribute__((ext_vector_type(2)));
typedef float v8f __attribute__((ext_vector_type(8)));

#define IN_DIM          1024
#define OUT_DIM         4096
#define BATCH           16384
#define ROWS_PER_BLOCK  16
#define KSEL            409            // int(4096 * 0.1)
#define PSTRIDE         (OUT_DIM + 4)  // LDS row padding (bank-conflict avoidance)
#define THREADS         256            // 8 wave32
#define NWAVES          8
#define CCAP            256            // candidate-compaction buffer per wave

// ---------------- pre-pass: "is x entirely non-negative?" flag ----------------
__global__ void init_flag_kernel(int* flag) {
    if (blockIdx.x == 0 && threadIdx.x == 0) flag[0] = 1; // assume all >= 0
}

__global__ void neg_scan_kernel(const float* __restrict__ x, int* flag, int n) {
    int stride = gridDim.x * blockDim.x * 4;
    bool anyNeg = false;
    for (int i = (blockIdx.x * blockDim.x + threadIdx.x) * 4; i < n; i += stride) {
        float4 v = *(const float4*)(x + i);   // n is a multiple of 4
        anyNeg |= (v.x < 0.f) | (v.y < 0.f) | (v.z < 0.f) | (v.w < 0.f);
    }
    if (__any(anyNeg) && (threadIdx.x & 31) == 0) atomicAnd(flag, 0);
}

// ordered-key decode: inverse of key = (u>>31) ? ~u : (u | 0x80000000)
__device__ __forceinline__ float kdec(unsigned k) {
    unsigned bits = (k & 0x80000000u) ? (k & 0x7FFFFFFFu) : ~k;
    return __uint_as_float(bits);
}

// ---------------- fused: GEMM (f32 WMMA) + top-K threshold + normalize ----------------
__global__ void __launch_bounds__(THREADS)
sps_fused_kernel(const float* __restrict__ x, const float* __restrict__ w,
                 const int* __restrict__ flag, float* __restrict__ out) {
    __shared__ unsigned p_lds[ROWS_PER_BLOCK * PSTRIDE]; // ~256 KB projected tile (f32 bits / keys)
    __shared__ unsigned kth[ROWS_PER_BLOCK];             // per-row K-th largest (key space)
    __shared__ float    sumsq[ROWS_PER_BLOCK];
    __shared__ float    rscale[ROWS_PER_BLOCK];
    __shared__ unsigned cbuf[NWAVES][CCAP];              // compacted select candidates
    __shared__ int      ccnt[NWAVES];

    const int tid  = threadIdx.x;
    const int lane = tid & 31;
    const int wave = tid >> 5;
    const int half = lane >> 4;      // 0: lanes 0-15, 1: lanes 16-31
    const int l16  = lane & 15;
    const int row0 = blockIdx.x * ROWS_PER_BLOCK;

    const bool bip = (flag[0] != 0); // x.min() >= 0  ->  use 2x-1 (wave-uniform)

    // ---------- Phase 1: projected[16][4096] = x_bip @ W^T via V_WMMA_F32_16X16X4_F32 ----------
    // A frag (16x4): lane(m=l16, half) holds A[m][half*2 + {0,1}]    -> float2 from x row
    // B frag (4x16): lane(n=l16, half) holds B[half*2 + {0,1}][n]    -> float2 from W row
    // 4 N-tiles per pass share each A fragment (1 A load : 4 WMMA).
    const float* Ap = x + (size_t)(row0 + l16) * IN_DIM + half * 2;
    for (int g = 0; g < 8; ++g) {                        // 8 groups x 4 tiles = wave's 32 tiles
        const int n0_0 = ((g * 4 + 0) * 8 + wave) * 16;
        const int n0_1 = ((g * 4 + 1) * 8 + wave) * 16;
        const int n0_2 = ((g * 4 + 2) * 8 + wave) * 16;
        const int n0_3 = ((g * 4 + 3) * 8 + wave) * 16;
        const float* Bp0 = w + (size_t)(n0_0 + l16) * IN_DIM + half * 2;
        const float* Bp1 = w + (size_t)(n0_1 + l16) * IN_DIM + half * 2;
        const float* Bp2 = w + (size_t)(n0_2 + l16) * IN_DIM + half * 2;
        const float* Bp3 = w + (size_t)(n0_3 + l16) * IN_DIM + half * 2;
        v8f a0 = {}, a1 = {}, a2 = {}, a3 = {};
        if (!bip) {
            #pragma unroll 8
            for (int k = 0; k < IN_DIM; k += 4) {
                v2f a = *(const v2f*)(Ap + k);
                a0 = __builtin_amdgcn_wmma_f32_16x16x4_f32(false, a, false, *(const v2f*)(Bp0 + k), (short)0, a0, false, false);
                a1 = __builtin_amdgcn_wmma_f32_16x16x4_f32(false, a, false, *(const v2f*)(Bp1 + k), (short)0, a1, false, false);
                a2 = __builtin_amdgcn_wmma_f32_16x16x4_f32(false, a, false, *(const v2f*)(Bp2 + k), (short)0, a2, false, false);
                a3 = __builtin_amdgcn_wmma_f32_16x16x4_f32(false, a, false, *(const v2f*)(Bp3 + k), (short)0, a3, false, false);
            }
        } else {
            #pragma unroll 8
            for (int k = 0; k < IN_DIM; k += 4) {
                v2f a = *(const v2f*)(Ap + k);
                a = a * 2.0f - 1.0f;                     // (x - 0.5) * 2
                a0 = __builtin_amdgcn_wmma_f32_16x16x4_f32(false, a, false, *(const v2f*)(Bp0 + k), (short)0, a0, false, false);
                a1 = __builtin_amdgcn_wmma_f32_16x16x4_f32(false, a, false, *(const v2f*)(Bp1 + k), (short)0, a1, false, false);
                a2 = __builtin_amdgcn_wmma_f32_16x16x4_f32(false, a, false, *(const v2f*)(Bp2 + k), (short)0, a2, false, false);
                a3 = __builtin_amdgcn_wmma_f32_16x16x4_f32(false, a, false, *(const v2f*)(Bp3 + k), (short)0, a3, false, false);
            }
        }
        // D layout: VGPR v, lanes 0-15 -> (M=v, N=lane); lanes 16-31 -> (M=v+8, N=lane-16)
        #pragma unroll
        for (int v = 0; v < 8; ++v) {
            const int rbase = (v + half * 8) * PSTRIDE + l16;
            p_lds[rbase + n0_0] = __float_as_uint(a0[v]);
            p_lds[rbase + n0_1] = __float_as_uint(a1[v]);
            p_lds[rbase + n0_2] = __float_as_uint(a2[v]);
            p_lds[rbase + n0_3] = __float_as_uint(a3[v]);
        }
    }
    if (tid < ROWS_PER_BLOCK) sumsq[tid] = 0.f;
    __syncthreads();

    // ---------- Phase 2a: transform tile to monotonic unsigned keys (once) ----------
    for (int idx = tid; idx < ROWS_PER_BLOCK * OUT_DIM; idx += THREADS) {
        const int rr = idx >> 12;
        const int cc = idx & (OUT_DIM - 1);
        unsigned u = p_lds[rr * PSTRIDE + cc];
        p_lds[rr * PSTRIDE + cc] = (u & 0x80000000u) ? ~u : (u | 0x80000000u);
    }
    __syncthreads();

    // ---------- Phase 2b: exact K-th largest key per row (MSB radix select + compaction) ----------
    for (int rr = 0; rr < 2; ++rr) {                     // wave handles rows {wave, wave+8}
        const int r = wave + rr * 8;
        const unsigned* krow = &p_lds[r * PSTRIDE];
        unsigned prefix = 0u;
        int  kneed = KSEL;
        int  cands = OUT_DIM;
        bool comp  = false;
        int  nbuf  = 0;
        for (int bit = 31; bit >= 0; --bit) {
            const unsigned bitv = 1u << bit;
            const unsigned hi   = (bit == 31) ? 0u : (0xFFFFFFFFu << (bit + 1));
            int cnt = 0;
            if (!comp) {
                for (int j = 0; j < OUT_DIM / 32; ++j) {          // stride-32: 1 bank/lane
                    unsigned key = krow[j * 32 + lane];
                    cnt += ((((key ^ prefix) & hi) == 0u) && (key & bitv)) ? 1 : 0;
                }
            } else {
                for (int j = lane; j < nbuf; j += 32) {
                    unsigned key = cbuf[wave][j];
                    cnt += ((((key ^ prefix) & hi) == 0u) && (key & bitv)) ? 1 : 0;
                }
            }
            for (int off = 16; off > 0; off >>= 1) cnt += __shfl_xor(cnt, off, 32);
            if (cnt >= kneed) { prefix |= bitv; cands = cnt; }
            else              { kneed -= cnt;   cands -= cnt; }
            // compact surviving candidates once they fit the per-wave buffer
            if (!comp && bit > 0 && cands <= CCAP) {
                if (lane == 0) ccnt[wave] = 0;
                const unsigned m = 0xFFFFFFFFu << bit;            // decided bits [31:bit]
                for (int j = 0; j < OUT_DIM / 32; ++j) {
                    unsigned key = krow[j * 32 + lane];
                    if (((key ^ prefix) & m) == 0u) {
                        int slot = atomicAdd(&ccnt[wave], 1);
                        cbuf[wave][slot] = key;
                    }
                }
                nbuf = cands;                                      // exact survivor count
                comp = true;
            }
        }
        if (lane == 0) kth[r] = prefix;                            // K-th largest, key space
    }
    __syncthreads();

    // ---------- Phase 3: encoded = relu(p) * (p >= t); row L2 norm ----------
    {
        const int r = tid >> 4;         // 16 threads per row
        const int g = tid & 15;
        const unsigned th = kth[r];
        const unsigned* krow = &p_lds[r * PSTRIDE];
        float s = 0.f;
        for (int j = g; j < OUT_DIM; j += 16) {
            unsigned key = krow[j];
            float e = (key >= th) ? fmaxf(kdec(key), 0.f) : 0.f;
            s += e * e;
        }
        atomicAdd(&sumsq[r], s);        // ds_add_f32
    }
    __syncthreads();
    if (tid < ROWS_PER_BLOCK)
        rscale[tid] = 1.0f / fmaxf(sqrtf(sumsq[tid]), 1e-12f);
    __syncthreads();

    // ---------- Phase 4: coalesced store ----------
    for (int idx = tid; idx < ROWS_PER_BLOCK * OUT_DIM; idx += THREADS) {
        const int rr = idx >> 12;               // / 4096
        const int cc = idx & (OUT_DIM - 1);
        unsigned key = p_lds[rr * PSTRIDE + cc];
        float e = (key >= kth[rr]) ? fmaxf(kdec(key), 0.f) : 0.f;
        out[(size_t)(row0 + rr) * OUT_DIM + cc] = e * rscale[rr];
    }
}

extern "C" void kernel_launch(void* const* d_in, const int* in_sizes, int n_in,
                              void* d_out, int out_size, void* d_ws, size_t ws_size,
                              hipStream_t stream) {
    const float* x = (const float*)d_in[0];            // (16384, 1024) f32
    const float* w = (const float*)d_in[1];            // (4096, 1024) f32
    float* out = (float*)d_out;                        // (16384, 4096) f32
    int* flag = (int*)d_ws;

    init_flag_kernel<<<1, 1, 0, stream>>>(flag);
    neg_scan_kernel<<<1024, 256, 0, stream>>>(x, flag, BATCH * IN_DIM);
    sps_fused_kernel<<<BATCH / ROWS_PER_BLOCK, THREADS, 0, stream>>>(x, w, flag, out);
}